// Encoder_MLP_24275155157245
// MI455X (gfx1250) — compile-verified
//
#include <hip/hip_runtime.h>
#include <hip/hip_bf16.h>

// Problem constants (from the reference): N=8, M=2048, D=128, L=128
#define NB   8
#define MM   2048
#define DD   128
#define LL   128
#define TWOL 256
#define ROWS (NB * MM)          // 16384 total rows

typedef __attribute__((ext_vector_type(2))) float v2f;
typedef __attribute__((ext_vector_type(4))) float v4f;
typedef __attribute__((ext_vector_type(8))) float v8f;

// --------------------------------------------------------------------------
// fp32 WMMA 16x16x4:  D = A(16x4) * B(4x16) + C(16x16)
// 8-arg pattern (probe-confirmed arity for _16x16x4): neg_a, A, neg_b, B,
// c_mod, C, reuse_a, reuse_b.
// --------------------------------------------------------------------------
__device__ __forceinline__ v8f wmma4(v2f a, v2f b, v8f c) {
  return __builtin_amdgcn_wmma_f32_16x16x4_f32(
      /*neg_a=*/false, a, /*neg_b=*/false, b,
      /*c_mod=*/(short)0, c, /*reuse_a=*/false, /*reuse_b=*/false);
}

// --------------------------------------------------------------------------
// Kernel 0: fold w1e (256x128) into w1e_sum (128x128):
//   concat(h,h) @ w1e  ==  h @ (w1e[:128] + w1e[128:])
// --------------------------------------------------------------------------
__global__ void prep_w1e(const float* __restrict__ w1e,
                         float* __restrict__ w1e_sum) {
  int i = blockIdx.x * blockDim.x + threadIdx.x;   // 0 .. 16383
  int k = i >> 7;
  int n = i & 127;
  w1e_sum[i] = w1e[k * LL + n] + w1e[(k + LL) * LL + n];
}

// --------------------------------------------------------------------------
// Kernel 1: fused node MLP + edge MLP (per 16-row tile).
// Block = 128 threads = 4 waves; each wave owns 16x16 output tiles.
// All GEMMs on the fp32 WMMA pipe (16x16x4), accumulate in fp32.
//
// fp32 A-fragment layout (ISA 7.12.2): lanes 0-15 -> K={k0,k0+1},
// lanes 16-31 -> K={k0+2,k0+3}; M = lane&15.
// fp32 B-fragment: same K split, N = lane&15.
// C/D: VGPR v -> M = v + 8*(lane>=16), N = lane&15.
// --------------------------------------------------------------------------
__global__ void __launch_bounds__(128)
mlp_kernel(const float* __restrict__ x,
           const float* __restrict__ w1n, const float* __restrict__ b1n,
           const float* __restrict__ w2n, const float* __restrict__ b2n,
           const float* __restrict__ w1e_sum, const float* __restrict__ b1e,
           const float* __restrict__ w2e, const float* __restrict__ b2e,
           float* __restrict__ out_h,      // ROWS x L
           float* __restrict__ e_ws) {     // ROWS
  __shared__ float t_s[16 * TWOL];   // ReLU(x@w1n+b1n) tile   (16 KB)
  __shared__ float h_s[16 * LL];     // h tile                 (8 KB)
  __shared__ float g_s[16 * LL];     // ReLU(h@w1e_sum+b1e)    (8 KB)

  const int tid  = threadIdx.x;
  const int wave = tid >> 5;
  const int lane = tid & 31;
  const int mlow = lane & 15;        // M index for A, N index for B/C
  const int hi8  = (lane >> 4) << 3; // 0 or 8   (C-matrix M offset)
  const int kb   = (lane >> 4) << 1; // 0 or 2   (A/B K offset within k0)
  const int row0 = blockIdx.x * 16;  // first global row of this tile

  // ---- Stage 1: t = ReLU(x_tile @ w1n + b1n)   (16 x 256) ----------------
  {
    const float* A = x + (size_t)row0 * DD;
    for (int nt = 0; nt < 4; ++nt) {
      const int n0 = wave * 64 + nt * 16;
      v8f acc = {};
      for (int k0 = 0; k0 < DD; k0 += 4) {
        v2f a, b;
        a.x = A[mlow * DD + k0 + kb];
        a.y = A[mlow * DD + k0 + kb + 1];
        b.x = w1n[(k0 + kb) * TWOL + n0 + mlow];
        b.y = w1n[(k0 + kb + 1) * TWOL + n0 + mlow];
        acc = wmma4(a, b, acc);
      }
      const float bias = b1n[n0 + mlow];
#pragma unroll
      for (int v = 0; v < 8; ++v) {
        float val = acc[v] + bias;
        t_s[(v + hi8) * TWOL + n0 + mlow] = val > 0.f ? val : 0.f;
      }
    }
  }
  __syncthreads();

  // ---- Stage 2: h = t @ w2n + b2n   (16 x 128), store to global + LDS ----
  {
    for (int nt = 0; nt < 2; ++nt) {
      const int n0 = wave * 32 + nt * 16;
      v8f acc = {};
      for (int k0 = 0; k0 < TWOL; k0 += 4) {
        v2f a, b;
        a.x = t_s[mlow * TWOL + k0 + kb];
        a.y = t_s[mlow * TWOL + k0 + kb + 1];
        b.x = w2n[(k0 + kb) * LL + n0 + mlow];
        b.y = w2n[(k0 + kb + 1) * LL + n0 + mlow];
        acc = wmma4(a, b, acc);
      }
      const float bias = b2n[n0 + mlow];
#pragma unroll
      for (int v = 0; v < 8; ++v) {
        float val = acc[v] + bias;
        const int mo = v + hi8;
        h_s[mo * LL + n0 + mlow] = val;
        out_h[(size_t)(row0 + mo) * LL + n0 + mlow] = val;
      }
    }
  }
  __syncthreads();

  // ---- Stage 3: g = ReLU(h @ w1e_sum + b1e)   (16 x 128) -----------------
  {
    for (int nt = 0; nt < 2; ++nt) {
      const int n0 = wave * 32 + nt * 16;
      v8f acc = {};
      for (int k0 = 0; k0 < LL; k0 += 4) {
        v2f a, b;
        a.x = h_s[mlow * LL + k0 + kb];
        a.y = h_s[mlow * LL + k0 + kb + 1];
        b.x = w1e_sum[(k0 + kb) * LL + n0 + mlow];
        b.y = w1e_sum[(k0 + kb + 1) * LL + n0 + mlow];
        acc = wmma4(a, b, acc);
      }
      const float bias = b1e[n0 + mlow];
#pragma unroll
      for (int v = 0; v < 8; ++v) {
        float val = acc[v] + bias;
        g_s[(v + hi8) * LL + n0 + mlow] = val > 0.f ? val : 0.f;
      }
    }
  }
  __syncthreads();

  // ---- Stage 4: e = (g @ w2e + b2e) / M   (16 x 1) -----------------------
  if (tid < 16) {
    float sum = 0.f;
#pragma unroll 8
    for (int k = 0; k < LL; ++k) sum += g_s[tid * LL + k] * w2e[k];
    e_ws[row0 + tid] = (sum + b2e[0]) * (1.0f / (float)MM);
  }
}

// --------------------------------------------------------------------------
// Kernel 2: edge[b,i,j] = e[b,i]  — 128 MiB pure streaming broadcast.
// One block per row; 256 threads x 2 non-temporal 16B stores = 2048 floats.
// NT hint: this output is never re-read, keep it out of the caches.
// --------------------------------------------------------------------------
__global__ void __launch_bounds__(256)
edge_bcast(const float* __restrict__ e, float* __restrict__ edge) {
  const int row = blockIdx.x;
  const float v = e[row];
  v4f vv = {v, v, v, v};
  v4f* base = (v4f*)(edge + (size_t)row * MM);
  __builtin_nontemporal_store(vv, base + threadIdx.x);
  __builtin_nontemporal_store(vv, base + threadIdx.x + 256);
}

// --------------------------------------------------------------------------
extern "C" void kernel_launch(void* const* d_in, const int* in_sizes, int n_in,
                              void* d_out, int out_size, void* d_ws, size_t ws_size,
                              hipStream_t stream) {
  (void)in_sizes; (void)n_in; (void)out_size; (void)ws_size;

  const float* x   = (const float*)d_in[0];
  const float* w1n = (const float*)d_in[1];
  const float* b1n = (const float*)d_in[2];
  const float* w2n = (const float*)d_in[3];
  const float* b2n = (const float*)d_in[4];
  const float* w1e = (const float*)d_in[5];
  const float* b1e = (const float*)d_in[6];
  const float* w2e = (const float*)d_in[7];
  const float* b2e = (const float*)d_in[8];

  float* out     = (float*)d_out;                   // [h | edge]
  float* out_h   = out;                             // ROWS * L
  float* out_edge= out + (size_t)ROWS * LL;         // N * M * M
  float* ws      = (float*)d_ws;
  float* w1e_sum = ws;                              // 128*128
  float* e_ws    = ws + LL * LL;                    // ROWS

  prep_w1e<<<dim3((LL * LL) / 256), dim3(256), 0, stream>>>(w1e, w1e_sum);

  mlp_kernel<<<dim3(ROWS / 16), dim3(128), 0, stream>>>(
      x, w1n, b1n, w2n, b2n, w1e_sum, b1e, w2e, b2e, out_h, e_ws);

  edge_bcast<<<dim3(ROWS), dim3(256), 0, stream>>>(e_ws, out_edge);
}